// LinearAggActor_8916352106905
// MI455X (gfx1250) — compile-verified
//
#include <hip/hip_runtime.h>
#include <math.h>

#define N_NODES 50000
#define FDIM    64
#define HDIM    16
#define NEDGE   1600000
#define ADIM    8
#define WAVES_PER_BLOCK 5          // 160 threads; 80 nodes/block; 50000/80 = 625 exact
#define TILE_BLOCKS 625

typedef __attribute__((ext_vector_type(2))) float v2f;
typedef __attribute__((ext_vector_type(8))) float v8f;

__device__ __forceinline__ v8f wmma_f32(v2f a, v2f b, v8f c) {
    // D = A(16x4) * B(4x16) + C(16x16), fp32
    return __builtin_amdgcn_wmma_f32_16x16x4_f32(
        /*neg_a=*/false, a, /*neg_b=*/false, b,
        /*c_mod=*/(short)0, c, /*reuse_a=*/false, /*reuse_b=*/false);
}

// Computes mlp(in[nb..nb+15, :]) into lds64 (16x64 tile, row-major).
// lds16 is 16x16 scratch. All lanes of all waves must participate (syncthreads inside).
__device__ __forceinline__ void mlp_tile(
    const float* __restrict__ in, int nb,
    const float* __restrict__ w1, const float* __restrict__ b1,
    const float* __restrict__ w2, const float* __restrict__ b2,
    const float* __restrict__ w3, const float* __restrict__ b3,
    float* lds16, float* lds64, int lane)
{
    const int m    = lane & 15;       // A-row / B-col / D-col index
    const int half = lane >> 4;
    const int koff = half * 2;        // A/B K sub-offset within a K=4 step
    const int n    = m;

    // ---- layer 1: [16x64] @ [64x16] ----
    v8f c1 = {};
    const float* xrow = in + (size_t)(nb + m) * FDIM;
#pragma unroll
    for (int ks = 0; ks < 16; ++ks) {
        const int k = ks * 4 + koff;
        v2f a; a.x = xrow[k];            a.y = xrow[k + 1];
        v2f b; b.x = w1[k * HDIM + n];   b.y = w1[(k + 1) * HDIM + n];
        c1 = wmma_f32(a, b, c1);
    }
    const float bias1 = b1[n];
#pragma unroll
    for (int j = 0; j < 8; ++j) {
        float v = c1[j] + bias1;
        lds16[(j + half * 8) * 16 + n] = v > 0.f ? v : 0.f;   // relu
    }
    __syncthreads();

    // ---- layer 2: [16x16] @ [16x16] ----
    v8f c2 = {};
#pragma unroll
    for (int ks = 0; ks < 4; ++ks) {
        const int k = ks * 4 + koff;
        v2f a; a.x = lds16[m * 16 + k];  a.y = lds16[m * 16 + k + 1];
        v2f b; b.x = w2[k * HDIM + n];   b.y = w2[(k + 1) * HDIM + n];
        c2 = wmma_f32(a, b, c2);
    }
    __syncthreads();   // everyone done reading lds16 before overwrite
    const float bias2 = b2[n];
#pragma unroll
    for (int j = 0; j < 8; ++j) {
        float v = c2[j] + bias2;
        lds16[(j + half * 8) * 16 + n] = v > 0.f ? v : 0.f;   // relu
    }
    __syncthreads();

    // ---- layer 3: [16x16] @ [16x64] -> lds64 ----
#pragma unroll
    for (int nt = 0; nt < 4; ++nt) {
        v8f c3 = {};
#pragma unroll
        for (int ks = 0; ks < 4; ++ks) {
            const int k = ks * 4 + koff;
            v2f a; a.x = lds16[m * 16 + k];            a.y = lds16[m * 16 + k + 1];
            v2f b; b.x = w3[k * FDIM + nt * 16 + n];   b.y = w3[(k + 1) * FDIM + nt * 16 + n];
            c3 = wmma_f32(a, b, c3);
        }
        const float bias3 = b3[nt * 16 + n];
#pragma unroll
        for (int j = 0; j < 8; ++j)
            lds64[(j + half * 8) * FDIM + nt * 16 + n] = c3[j] + bias3;  // linear
    }
    __syncthreads();
}

// y = mlp(x)
__global__ __launch_bounds__(32 * WAVES_PER_BLOCK)
void mlp_store_kernel(const float* __restrict__ x,
                      const float* __restrict__ w1, const float* __restrict__ b1,
                      const float* __restrict__ w2, const float* __restrict__ b2,
                      const float* __restrict__ w3, const float* __restrict__ b3,
                      float* __restrict__ y)
{
    __shared__ float lds16_s[WAVES_PER_BLOCK * 16 * 16];
    __shared__ float lds64_s[WAVES_PER_BLOCK * 16 * 64];
    const int wave = threadIdx.x >> 5, lane = threadIdx.x & 31;
    const int nb = (blockIdx.x * WAVES_PER_BLOCK + wave) * 16;
    float* lds16 = lds16_s + wave * 256;
    float* lds64 = lds64_s + wave * 1024;

    mlp_tile(x, nb, w1, b1, w2, b2, w3, b3, lds16, lds64, lane);

    float4* dst = (float4*)(y + (size_t)nb * FDIM);
    const float4* srcv = (const float4*)lds64;
#pragma unroll
    for (int i = 0; i < 8; ++i) dst[i * 32 + lane] = srcv[i * 32 + lane];
}

// h[nb..nb+15, :] += mlp(a_in tile) @ fc1_wk[64x64]
__global__ __launch_bounds__(32 * WAVES_PER_BLOCK)
void field_fc1_kernel(const float* __restrict__ a_in,
                      const float* __restrict__ w1, const float* __restrict__ b1,
                      const float* __restrict__ w2, const float* __restrict__ b2,
                      const float* __restrict__ w3, const float* __restrict__ b3,
                      const float* __restrict__ fc1_wk,
                      float* __restrict__ h)
{
    __shared__ float lds16_s[WAVES_PER_BLOCK * 16 * 16];
    __shared__ float lds64_s[WAVES_PER_BLOCK * 16 * 64];
    const int wave = threadIdx.x >> 5, lane = threadIdx.x & 31;
    const int nb = (blockIdx.x * WAVES_PER_BLOCK + wave) * 16;
    float* lds16 = lds16_s + wave * 256;
    float* lds64 = lds64_s + wave * 1024;

    mlp_tile(a_in, nb, w1, b1, w2, b2, w3, b3, lds16, lds64, lane);

    const int m = lane & 15, half = lane >> 4, koff = half * 2, n = m;
#pragma unroll
    for (int nt = 0; nt < 4; ++nt) {
        v8f ch;
#pragma unroll
        for (int j = 0; j < 8; ++j)
            ch[j] = h[(size_t)(nb + j + half * 8) * FDIM + nt * 16 + n];
#pragma unroll
        for (int ks = 0; ks < 16; ++ks) {
            const int k = ks * 4 + koff;
            v2f a; a.x = lds64[m * FDIM + k];              a.y = lds64[m * FDIM + k + 1];
            v2f b; b.x = fc1_wk[k * FDIM + nt * 16 + n];   b.y = fc1_wk[(k + 1) * FDIM + nt * 16 + n];
            ch = wmma_f32(a, b, ch);
        }
#pragma unroll
        for (int j = 0; j < 8; ++j)
            h[(size_t)(nb + j + half * 8) * FDIM + nt * 16 + n] = ch[j];
    }
}

__global__ void h_init_kernel(float* __restrict__ h, const float* __restrict__ fc1_b)
{
    const int i = blockIdx.x * blockDim.x + threadIdx.x;
    if (i < N_NODES * FDIM) h[i] = fc1_b[i & (FDIM - 1)];
}

__global__ void count_kernel(const long long* __restrict__ ei, float* __restrict__ cnt)
{
    const int e = blockIdx.x * blockDim.x + threadIdx.x;
    if (e < NEDGE) atomicAdd(&cnt[(int)ei[NEDGE + e]], 1.0f);
}

// scatter-sum: 8 threads per edge, 8 features (2x float4) each
__global__ void scatter_kernel(const float* __restrict__ src_feat,
                               const long long* __restrict__ ei,
                               float* __restrict__ acc)
{
    const int tid = blockIdx.x * blockDim.x + threadIdx.x;
    if (tid >= NEDGE * 8) return;
    const int e = tid >> 3;
    const int c = (tid & 7) * 8;
    const int s = (int)ei[e];
    const int d = (int)ei[NEDGE + e];
    const float4* sp = (const float4*)(src_feat + (size_t)s * FDIM + c);
    const float4 v0 = sp[0], v1 = sp[1];
    float* dp = acc + (size_t)d * FDIM + c;
    atomicAdd(dp + 0, v0.x); atomicAdd(dp + 1, v0.y);
    atomicAdd(dp + 2, v0.z); atomicAdd(dp + 3, v0.w);
    atomicAdd(dp + 4, v1.x); atomicAdd(dp + 5, v1.y);
    atomicAdd(dp + 6, v1.z); atomicAdd(dp + 7, v1.w);
}

__global__ void normalize_kernel(float* __restrict__ buf, const float* __restrict__ cnt)
{
    const int i = blockIdx.x * blockDim.x + threadIdx.x;
    if (i < N_NODES * FDIM) {
        const float c = cnt[i >> 6];
        buf[i] = buf[i] / fmaxf(c, 1.0f);
    }
}

__global__ void softmax_kernel(const float* __restrict__ h,
                               const float* __restrict__ fc2_w,
                               const float* __restrict__ fc2_b,
                               float* __restrict__ out)
{
    const int nidx = blockIdx.x * blockDim.x + threadIdx.x;
    if (nidx >= N_NODES) return;
    float acc[ADIM];
#pragma unroll
    for (int a = 0; a < ADIM; ++a) acc[a] = fc2_b[a];
    const float* hr = h + (size_t)nidx * FDIM;
#pragma unroll 4
    for (int j = 0; j < FDIM; ++j) {
        const float hv = hr[j];
#pragma unroll
        for (int a = 0; a < ADIM; ++a) acc[a] += hv * fc2_w[j * ADIM + a];
    }
    float mx = acc[0];
#pragma unroll
    for (int a = 1; a < ADIM; ++a) mx = fmaxf(mx, acc[a]);
    float sum = 0.f;
#pragma unroll
    for (int a = 0; a < ADIM; ++a) { acc[a] = expf(acc[a] - mx); sum += acc[a]; }
    const float inv = 1.0f / sum;
#pragma unroll
    for (int a = 0; a < ADIM; ++a) out[(size_t)nidx * ADIM + a] = acc[a] * inv;
}

extern "C" void kernel_launch(void* const* d_in, const int* in_sizes, int n_in,
                              void* d_out, int out_size, void* d_ws, size_t ws_size,
                              hipStream_t stream)
{
    const float*     x     = (const float*)d_in[0];
    const long long* ei    = (const long long*)d_in[1];
    const float*     w1    = (const float*)d_in[2];
    const float*     b1    = (const float*)d_in[3];
    const float*     w2    = (const float*)d_in[4];
    const float*     b2    = (const float*)d_in[5];
    const float*     w3    = (const float*)d_in[6];
    const float*     b3    = (const float*)d_in[7];
    const float*     fc1_w = (const float*)d_in[8];
    const float*     fc1_b = (const float*)d_in[9];
    const float*     fc2_w = (const float*)d_in[10];
    const float*     fc2_b = (const float*)d_in[11];

    const size_t NF = (size_t)N_NODES * FDIM;
    float* ws   = (float*)d_ws;
    float* y    = ws;            // mlp(x) == a_0
    float* bufA = ws + NF;
    float* bufB = ws + 2 * NF;
    float* h    = ws + 3 * NF;   // fc1 accumulator, N x 64
    float* cnt  = ws + 4 * NF;   // in-degree, N

    // in-degree (same for every agg step)
    hipMemsetAsync(cnt, 0, N_NODES * sizeof(float), stream);
    count_kernel<<<(NEDGE + 255) / 256, 256, 0, stream>>>(ei, cnt);

    // y = mlp(x)
    mlp_store_kernel<<<TILE_BLOCKS, 32 * WAVES_PER_BLOCK, 0, stream>>>(
        x, w1, b1, w2, b2, w3, b3, y);

    // h = fc1_b broadcast, then accumulate field_k @ fc1_w_k
    h_init_kernel<<<(int)((NF + 255) / 256), 256, 0, stream>>>(h, fc1_b);

    // k = 0: field_0 = mlp(y)
    field_fc1_kernel<<<TILE_BLOCKS, 32 * WAVES_PER_BLOCK, 0, stream>>>(
        y, w1, b1, w2, b2, w3, b3, fc1_w, h);

    const float* cur = y;
    for (int k = 1; k <= 7; ++k) {
        float* nxt = (k & 1) ? bufA : bufB;
        hipMemsetAsync(nxt, 0, NF * sizeof(float), stream);
        scatter_kernel<<<(NEDGE * 8 + 255) / 256, 256, 0, stream>>>(cur, ei, nxt);
        normalize_kernel<<<(int)((NF + 255) / 256), 256, 0, stream>>>(nxt, cnt);
        field_fc1_kernel<<<TILE_BLOCKS, 32 * WAVES_PER_BLOCK, 0, stream>>>(
            nxt, w1, b1, w2, b2, w3, b3, fc1_w + (size_t)k * FDIM * FDIM, h);
        cur = nxt;
    }

    softmax_kernel<<<(N_NODES + 255) / 256, 256, 0, stream>>>(
        h, fc2_w, fc2_b, (float*)d_out);
}